// Mamba_81913616269596
// MI455X (gfx1250) — compile-verified
//
#include <hip/hip_runtime.h>
#include <hip/hip_bf16.h>
#include <math.h>

typedef float v2f __attribute__((ext_vector_type(2)));
typedef float v8f __attribute__((ext_vector_type(8)));
typedef int   i32x4 __attribute__((vector_size(16)));

#if defined(__has_builtin)
#if __has_builtin(__builtin_amdgcn_global_load_async_to_lds_b128)
#define HAVE_ASYNC_LDS 1
#endif
#endif

#ifdef HAVE_ASYNC_LDS
__device__ __forceinline__ void async_cp16(const float* g, float* l) {
    // param0: vector-of-4-int pointer (global src), param1: LDS dst, imm offset, imm cpol
    __builtin_amdgcn_global_load_async_to_lds_b128((i32x4*)g, (i32x4*)l, 0, 0);
}
__device__ __forceinline__ void wait_async0() {
#if __has_builtin(__builtin_amdgcn_s_wait_asynccnt)
    __builtin_amdgcn_s_wait_asynccnt(0);
#else
    asm volatile("s_wait_asynccnt 0" ::: "memory");
#endif
}
#endif

namespace {
constexpr int kB = 2, kL = 1024, kDM = 512, kDI = 1024, kH = 8, kN = 16, kHD = 128;
constexpr int kDTR = 32, kPROJ = 560, kXZC = 2048, kM = kB * kL;   // 2048 rows
constexpr int TS = 36;   // tile k-stride (32 + pad): b64-aligned, conflict-free
}

__device__ __forceinline__ float sigmoid_f(float x) { return 1.f / (1.f + __expf(-x)); }

// C[M x Nout] = act( A[M x K](row stride lda) @ W[Nout x K]^T + bias )
// Block: 256 threads = 8 waves; block tile 128x64; K staged in 32-chunks via LDS
// (async global->LDS b128 when available). Both LDS tiles are fragment-major
// [row][k] with stride 36 so every WMMA fragment is one ds_load_b64.
__global__ __launch_bounds__(256)
void gemm_wmma_f32(const float* __restrict__ A, int lda,
                   const float* __restrict__ W,
                   const float* __restrict__ bias,
                   float* __restrict__ C,
                   int K, int Nout, int act)
{
    __shared__ float As[128 * TS];
    __shared__ float Bs[64 * TS];

    const int nBlocks = (Nout + 63) >> 6;
    const int bm = (blockIdx.x / nBlocks) * 128;
    const int bn = (blockIdx.x % nBlocks) * 64;
    const int tid  = threadIdx.x;
    const int wave = tid >> 5;
    const int lane = tid & 31;
    const int l16  = lane & 15;
    const int hi   = lane >> 4;       // wave half: K split per ISA f32 A layout

    v8f acc[4];
    #pragma unroll
    for (int t = 0; t < 4; ++t)
        #pragma unroll
        for (int r = 0; r < 8; ++r) acc[t][r] = 0.f;

    const int arow_base = tid >> 1;            // A staging: 2 threads/row
    const int akb       = (tid & 1) << 4;
    const int bnn       = tid >> 3;            // B staging: 8 threads/row
    const int bkk       = (tid & 7) << 2;

    for (int k0 = 0; k0 < K; k0 += 32) {
#ifdef HAVE_ASYNC_LDS
        {   // A tile 128x32: 4 async b128 per thread
            const float* ag = A + (size_t)(bm + arow_base) * lda + (k0 + akb);
            float* as = &As[arow_base * TS + akb];
            async_cp16(ag + 0,  as + 0);
            async_cp16(ag + 4,  as + 4);
            async_cp16(ag + 8,  as + 8);
            async_cp16(ag + 12, as + 12);
        }
        {   // B tile 64x32 (fragment-major, same layout as A): 2 async b128
            #pragma unroll
            for (int pass = 0; pass < 2; ++pass) {
                const int nn   = bnn + (pass << 5);             // 0..63
                const int nsrc = (bn + nn < Nout) ? (bn + nn) : (Nout - 1);
                async_cp16(W + (size_t)nsrc * K + (k0 + bkk),
                           &Bs[nn * TS + bkk]);
            }
        }
        wait_async0();
        __syncthreads();
#else
        {
            const float* ag = A + (size_t)(bm + arow_base) * lda + (k0 + akb);
            float* as = &As[arow_base * TS + akb];
            ((float4*)as)[0] = ((const float4*)ag)[0];
            ((float4*)as)[1] = ((const float4*)ag)[1];
            ((float4*)as)[2] = ((const float4*)ag)[2];
            ((float4*)as)[3] = ((const float4*)ag)[3];
        }
        #pragma unroll
        for (int pass = 0; pass < 2; ++pass) {
            const int nn = bnn + (pass << 5);
            float4 wv = make_float4(0.f, 0.f, 0.f, 0.f);
            if (bn + nn < Nout)
                wv = *(const float4*)(W + (size_t)(bn + nn) * K + (k0 + bkk));
            *(float4*)&Bs[nn * TS + bkk] = wv;
        }
        __syncthreads();
#endif

        // --- hoist all fragments (one ds_load_b64 each), then 32 chained WMMAs
        const int abase = (wave * 16 + l16) * TS + hi * 2;
        const int bbase = l16 * TS + hi * 2;
        v2f afrag[8];
        v2f bfrag[4][8];
        #pragma unroll
        for (int j = 0; j < 8; ++j)
            afrag[j] = *(const v2f*)&As[abase + 4 * j];
        #pragma unroll
        for (int t = 0; t < 4; ++t)
            #pragma unroll
            for (int j = 0; j < 8; ++j)
                bfrag[t][j] = *(const v2f*)&Bs[bbase + t * 16 * TS + 4 * j];

        #pragma unroll
        for (int j = 0; j < 8; ++j)
            #pragma unroll
            for (int t = 0; t < 4; ++t)
                acc[t] = __builtin_amdgcn_wmma_f32_16x16x4_f32(
                    false, afrag[j], false, bfrag[t][j], (short)0, acc[t],
                    false, false);
        __syncthreads();
    }

    // --- epilogue: bias + optional softplus, C/D layout rows r / r+8 per half
    #pragma unroll
    for (int t = 0; t < 4; ++t) {
        const int col = bn + t * 16 + l16;
        if (col >= Nout) continue;
        const float bv = bias ? bias[col] : 0.f;
        #pragma unroll
        for (int r = 0; r < 8; ++r) {
            const int row = bm + wave * 16 + hi * 8 + r;
            float v = acc[t][r] + bv;
            if (act == 1) v = (v > 15.f) ? v : log1pf(__expf(v));
            C[(size_t)row * Nout + col] = v;
        }
    }
}

// Causal depthwise conv (K=3) over x_proj = xz[:, :, :d_inner], then SiLU.
__global__ __launch_bounds__(256)
void conv_silu(const float* __restrict__ xz, const float* __restrict__ cw,
               const float* __restrict__ cb, float* __restrict__ xc)
{
    const int idx = blockIdx.x * 256 + threadIdx.x;
    if (idx >= kM * kDI) return;
    const int c = idx & (kDI - 1);
    const int m = idx >> 10;
    const int t = m & (kL - 1);
    const int b = m >> 10;
    float acc = cb[c];
    #pragma unroll
    for (int k = 0; k < 3; ++k) {
        const int tt = t + k - 2;
        if (tt >= 0)
            acc = fmaf(xz[(size_t)(b * kL + tt) * kXZC + c], cw[c * 3 + k], acc);
    }
    xc[idx] = acc * sigmoid_f(acc);
}

// Fused complex SSM recurrence + output gating.
// One wave handles (b, h, d0, d0+1): lanes 0-15 -> (d0, n=lane), 16-31 -> (d0+1, n).
// State h = alpha*h + u carried in registers across L; N-reduction via shfl_xor
// within 16-lane halves (wave32). Writes yz = (sum_n h*C + D*xc) * silu(z).
__global__ __launch_bounds__(256)
void scan_fused(const float* __restrict__ proj, const float* __restrict__ delta,
                const float* __restrict__ xc, const float* __restrict__ xz,
                const float* __restrict__ A_log, const float* __restrict__ A_imag,
                const float* __restrict__ Dp, float* __restrict__ yz)
{
    const int wid   = blockIdx.x * 8 + (threadIdx.x >> 5);  // 0..1023
    const int lane  = threadIdx.x & 31;
    const int n     = lane & 15;
    const int dpair = wid & 63;
    const int h     = (wid >> 6) & 7;
    const int b     = wid >> 9;
    const int d     = dpair * 2 + (lane >> 4);
    const int c     = h * kHD + d;

    const float Are = -__expf(A_log[h * kN + n]);
    const float Aim = A_imag[h * kN + n];
    const float Dv  = Dp[c];

    float hre = 0.f, him = 0.f, bxre_p = 0.f, bxim_p = 0.f;

    for (int t = 0; t < kL; ++t) {
        const size_t m = (size_t)b * kL + t;
        const float* pr = proj + m * kPROJ;
        const float dl  = delta[m * kDI + c];
        const float Bre = pr[kDTR + 0 * kH * kN + h * kN + n];
        const float Bim = pr[kDTR + 1 * kH * kN + h * kN + n];
        const float Cre = pr[kDTR + 2 * kH * kN + h * kN + n];
        const float Cim = pr[kDTR + 3 * kH * kN + h * kN + n];
        const float lg  = sigmoid_f(pr[kDTR + 4 * kH * kN + h]);
        const float eg  = sigmoid_f(pr[kDTR + 4 * kH * kN + kH + h]);
        const float xv  = xc[m * kDI + c];

        float dtre = dl * Are;
        dtre = fminf(fmaxf(dtre, -20.f), 20.f);
        const float dtim = dl * Aim;
        const float mag  = __expf(dtre);
        float sn, cs;
        __sincosf(dtim, &sn, &cs);
        const float are = mag * cs, aim = mag * sn;

        const float bxre = xv * Bre, bxim = xv * Bim;
        const float bre = (1.f - lg) * dl * are;
        const float bim = (1.f - lg) * dl * aim;
        const float gam = lg * dl * eg;
        const float ure = bre * bxre_p - bim * bxim_p + gam * bxre;
        const float uim = bre * bxim_p + bim * bxre_p + gam * bxim;
        const float nre = are * hre - aim * him + ure;
        const float nim = are * him + aim * hre + uim;
        hre = nre; him = nim;
        bxre_p = bxre; bxim_p = bxim;

        float contrib = hre * Cre + him * Cim;
        contrib += __shfl_xor(contrib, 8, 32);
        contrib += __shfl_xor(contrib, 4, 32);
        contrib += __shfl_xor(contrib, 2, 32);
        contrib += __shfl_xor(contrib, 1, 32);
        if (n == 0) {
            const float zr = xz[m * kXZC + kDI + c];
            const float y  = contrib + Dv * xv;
            yz[m * kDI + c] = y * (zr * sigmoid_f(zr));
        }
    }
}

extern "C" void kernel_launch(void* const* d_in, const int* in_sizes, int n_in,
                              void* d_out, int out_size, void* d_ws, size_t ws_size,
                              hipStream_t stream)
{
    (void)in_sizes; (void)n_in; (void)out_size; (void)ws_size;
    const float* x      = (const float*)d_in[0];
    const float* W_in   = (const float*)d_in[1];
    const float* conv_w = (const float*)d_in[2];
    const float* conv_b = (const float*)d_in[3];
    const float* W_x    = (const float*)d_in[4];
    const float* W_dt   = (const float*)d_in[5];
    const float* b_dt   = (const float*)d_in[6];
    const float* A_log  = (const float*)d_in[7];
    const float* A_imag = (const float*)d_in[8];
    const float* Dp     = (const float*)d_in[9];
    const float* W_out  = (const float*)d_in[10];
    float* out = (float*)d_out;

    float* xzb   = (float*)d_ws;                         // 2048 x 2048
    float* xcb   = xzb   + (size_t)kM * kXZC;            // 2048 x 1024
    float* projb = xcb   + (size_t)kM * kDI;             // 2048 x 560
    float* deltb = projb + (size_t)kM * kPROJ;           // 2048 x 1024
    float* yzb   = deltb + (size_t)kM * kDI;             // 2048 x 1024

    // 1) xz = x @ W_in^T
    gemm_wmma_f32<<<(kM / 128) * (kXZC / 64), 256, 0, stream>>>(
        x, kDM, W_in, nullptr, xzb, kDM, kXZC, 0);
    // 2) causal depthwise conv + SiLU -> xc
    conv_silu<<<(kM * kDI) / 256, 256, 0, stream>>>(xzb, conv_w, conv_b, xcb);
    // 3) proj = xc @ W_x^T   (Nout = 560, masked edge tiles)
    gemm_wmma_f32<<<(kM / 128) * ((kPROJ + 63) / 64), 256, 0, stream>>>(
        xcb, kDI, W_x, nullptr, projb, kDI, kPROJ, 0);
    // 4) delta = softplus(proj[:, :32] @ W_dt^T + b_dt)  (fused bias+softplus)
    gemm_wmma_f32<<<(kM / 128) * (kDI / 64), 256, 0, stream>>>(
        projb, kPROJ, W_dt, b_dt, deltb, kDTR, kDI, 1);
    // 5) fused complex SSM scan + gating -> yz
    scan_fused<<<128, 256, 0, stream>>>(projb, deltb, xcb, xzb,
                                        A_log, A_imag, Dp, yzb);
    // 6) out = yz @ W_out^T
    gemm_wmma_f32<<<(kM / 128) * (kDM / 64), 256, 0, stream>>>(
        yzb, kDI, W_out, nullptr, out, kDI, kDM, 0);
}